// S4Block_10393820856519
// MI455X (gfx1250) — compile-verified
//
#include <hip/hip_runtime.h>
#include <math.h>
#include <stdint.h>

#define L_SEQ 2048
#define NFFT  4096
#define NST   64
#define DM    512
#define BATCH 8
#define TWO_PI 6.28318530717958647692f

typedef __attribute__((ext_vector_type(2))) float v2f;
typedef __attribute__((ext_vector_type(8))) float v8f;

// ---------- complex helpers ----------
__device__ __forceinline__ float2 cmul(float2 a, float2 b) {
    return make_float2(a.x * b.x - a.y * b.y, a.x * b.y + a.y * b.x);
}
__device__ __forceinline__ float2 cconj(float2 a) { return make_float2(a.x, -a.y); }
__device__ __forceinline__ float2 cfma2(float2 a, float2 b, float2 c) {
    return make_float2(fmaf(a.x, b.x, fmaf(-a.y, b.y, c.x)),
                       fmaf(a.x, b.y, fmaf(a.y, b.x, c.y)));
}

// ---------- in-LDS iterative radix-2 DIT FFT ----------
// sign = -1 forward (e^{-2pi i}), +1 inverse (unnormalized)
__device__ void fft_lds(float2* s, int n, int logn, float sign) {
    const int tid = threadIdx.x;
    const int nt  = blockDim.x;
    for (int i = tid; i < n; i += nt) {
        unsigned r = __brev((unsigned)i) >> (32 - logn);
        if (r > (unsigned)i) { float2 t = s[i]; s[i] = s[r]; s[r] = t; }
    }
    __syncthreads();
    for (int st = 1; st <= logn; ++st) {
        const int half = 1 << (st - 1);
        const float ang = sign * TWO_PI / (float)(1 << st);
        for (int j = tid; j < (n >> 1); j += nt) {
            const int k  = j & (half - 1);
            const int i0 = ((j >> (st - 1)) << st) | k;
            const int i1 = i0 + half;
            float sn, cs;
            __sincosf(ang * (float)k, &sn, &cs);
            float2 a = s[i0], b = s[i1];
            float tr = cs * b.x - sn * b.y;
            float ti = cs * b.y + sn * b.x;
            s[i0] = make_float2(a.x + tr, a.y + ti);
            s[i1] = make_float2(a.x - tr, a.y - ti);
        }
        __syncthreads();
    }
}

// ---------- kernel 1: build Kf[h, 0..4095] ----------
__global__ __launch_bounds__(256) void kgen_kernel(
    const float* __restrict__ lam_re, const float* __restrict__ lam_im,
    const float* __restrict__ p_re,   const float* __restrict__ p_im,
    const float* __restrict__ B_re,   const float* __restrict__ B_im,
    const float* __restrict__ Ct_re,  const float* __restrict__ Ct_im,
    const float* __restrict__ log_step, float2* __restrict__ Kf) {
    __shared__ float2 s[NFFT];
    __shared__ float2 lamS[NST], w00[NST], w01[NST], w10[NST], w11[NST];
    const int h = blockIdx.x;
    const int tid = threadIdx.x;

    if (tid < NST) {
        int n = tid;
        float2 lam = make_float2(lam_re[n], lam_im[n]);
        float2 p   = make_float2(p_re[n],   p_im[n]);
        float2 Bc  = make_float2(B_re[h * NST + n],  B_im[h * NST + n]);
        float2 Ct  = make_float2(Ct_re[h * NST + n], Ct_im[h * NST + n]);
        lamS[n] = lam;
        w00[n] = cmul(cconj(Ct), Bc);
        w01[n] = cmul(cconj(Ct), p);
        w10[n] = cmul(cconj(p),  Bc);
        w11[n] = cmul(cconj(p),  p);
    }
    __syncthreads();

    const float step = expf(log_step[h]);
    const float two_over_step = 2.0f / step;

    for (int l = tid; l < L_SEQ; l += 256) {
        float sn, cs;
        __sincosf(-TWO_PI * (float)l / (float)L_SEQ, &sn, &cs);
        float2 num = make_float2(1.0f - cs, -sn);   // 1 - omega
        float2 den = make_float2(1.0f + cs,  sn);   // 1 + omega
        float id = 1.0f / (den.x * den.x + den.y * den.y);
        float2 ratio = cmul(num, cconj(den));
        ratio.x *= id; ratio.y *= id;
        float2 g = make_float2(ratio.x * two_over_step, ratio.y * two_over_step);
        float2 c = make_float2(2.0f * den.x * id, -2.0f * den.y * id);

        float2 k00 = make_float2(0, 0), k01 = make_float2(0, 0);
        float2 k10 = make_float2(0, 0), k11 = make_float2(0, 0);
        #pragma unroll 4
        for (int n = 0; n < NST; ++n) {
            float2 d = make_float2(g.x - lamS[n].x, g.y - lamS[n].y);
            float idn = 1.0f / (d.x * d.x + d.y * d.y);
            float2 cau = make_float2(d.x * idn, -d.y * idn);
            k00 = cfma2(w00[n], cau, k00);
            k01 = cfma2(w01[n], cau, k01);
            k10 = cfma2(w10[n], cau, k10);
            k11 = cfma2(w11[n], cau, k11);
        }
        float2 opk = make_float2(1.0f + k11.x, k11.y);
        float iopk = 1.0f / (opk.x * opk.x + opk.y * opk.y);
        float2 invopk = make_float2(opk.x * iopk, -opk.y * iopk);
        float2 t = cmul(k01, cmul(invopk, k10));
        float2 inner = make_float2(k00.x - t.x, k00.y - t.y);
        s[l] = cmul(c, inner);
    }
    __syncthreads();

    fft_lds(s, L_SEQ, 11, +1.0f);   // ifft(at_roots)
    float kt[8];
    #pragma unroll
    for (int i = 0; i < 8; ++i) kt[i] = s[tid + i * 256].x * (1.0f / (float)L_SEQ);
    __syncthreads();
    #pragma unroll
    for (int i = 0; i < 8; ++i) {
        s[tid + i * 256]         = make_float2(kt[i], 0.0f);
        s[L_SEQ + tid + i * 256] = make_float2(0.0f, 0.0f);
    }
    __syncthreads();
    fft_lds(s, NFFT, 12, -1.0f);    // rfft(K, 4096) (full spectrum)
    for (int f = tid; f < NFFT; f += 256) Kf[(size_t)h * NFFT + f] = s[f];
}

// ---------- kernel 2: FFT convolution + D*u + GELU -> z ----------
// Kf row is async-copied into LDS while the forward FFT of u runs.
__global__ __launch_bounds__(256) void conv_kernel(
    const float* __restrict__ u, const float* __restrict__ Dvec,
    const float2* __restrict__ Kf, float* __restrict__ zbuf) {
    __shared__ float2 s[NFFT];    // 32 KB
    __shared__ float2 kfS[NFFT];  // 32 KB (async-filled)
    const int bh = blockIdx.x;
    const int b  = bh >> 9;       // / 512
    const int h  = bh & (DM - 1); // % 512
    const int tid = threadIdx.x;
    const float* ub = u + (size_t)b * L_SEQ * DM + h;

    for (int t = tid; t < L_SEQ; t += 256)
        s[t] = make_float2(ub[(size_t)t * DM], 0.0f);
    for (int t = L_SEQ + tid; t < NFFT; t += 256)
        s[t] = make_float2(0.0f, 0.0f);

    // Kick off async Kf -> LDS copy (ASYNCcnt-tracked); hidden under the FFT.
    {
        const unsigned ldsBase = (unsigned)(uintptr_t)(&kfS[0]);
        const float2* kfG = Kf + (size_t)h * NFFT;
        #pragma unroll
        for (int i = 0; i < 8; ++i) {
            const int g = tid + i * 256;            // 16-byte granule index
            unsigned lds_addr = ldsBase + (unsigned)g * 16u;
            unsigned long long gaddr = (unsigned long long)(uintptr_t)(kfG + (size_t)g * 2);
            asm volatile("global_load_async_to_lds_b128 %0, %1, off"
                         :: "v"(lds_addr), "v"(gaddr) : "memory");
        }
    }
    __syncthreads();

    fft_lds(s, NFFT, 12, -1.0f);

    asm volatile("s_wait_asynccnt 0x0" ::: "memory");
    __syncthreads();

    for (int f = tid; f < NFFT; f += 256) {
        float2 a = s[f], k = kfS[f];
        s[f] = make_float2(a.x * k.x - a.y * k.y, a.x * k.y + a.y * k.x);
    }
    __syncthreads();

    fft_lds(s, NFFT, 12, +1.0f);

    const float dh = Dvec[h];
    for (int t = tid; t < L_SEQ; t += 256) {
        float y = s[t].x * (1.0f / (float)NFFT) + dh * ub[(size_t)t * DM];
        float z = 0.5f * y * (1.0f + erff(y * 0.70710678118654752440f));
        zbuf[((size_t)b * L_SEQ + t) * DM + h] = z;
    }
}

// ---------- kernel 3: fp32 WMMA GEMM  r = z @ W^T + b + u ----------
// Each wave computes a 16x32 output tile: one A fragment feeds two WMMAs.
__global__ __launch_bounds__(256) void gemm_kernel(
    const float* __restrict__ z, const float* __restrict__ W,
    const float* __restrict__ bias, const float* __restrict__ u,
    float* __restrict__ out) {
    const int lane = threadIdx.x & 31;
    const int wave = threadIdx.x >> 5;
    const int tile = blockIdx.x * 8 + wave;      // 16x32 tiles
    const int nTiles = DM / 32;                  // 16
    const int mBase = (tile / nTiles) * 16;
    const int oBase = (tile % nTiles) * 32;
    const int lane16 = lane & 15;
    const int kHi = (lane >> 4) * 2;             // lanes 16-31 hold K=2,3

    const float* aRow  = z + (size_t)(mBase + lane16) * DM + kHi;
    const float* bRow0 = W + (size_t)(oBase + lane16) * DM + kHi;
    const float* bRow1 = W + (size_t)(oBase + 16 + lane16) * DM + kHi;

    v8f acc0 = {0.f, 0.f, 0.f, 0.f, 0.f, 0.f, 0.f, 0.f};
    v8f acc1 = {0.f, 0.f, 0.f, 0.f, 0.f, 0.f, 0.f, 0.f};
    #pragma unroll 8
    for (int kk = 0; kk < DM; kk += 4) {
        v2f a  = *(const v2f*)(aRow  + kk);
        v2f b0 = *(const v2f*)(bRow0 + kk);
        v2f b1 = *(const v2f*)(bRow1 + kk);
        acc0 = __builtin_amdgcn_wmma_f32_16x16x4_f32(
            false, a, false, b0, (short)0, acc0, false, false);
        acc1 = __builtin_amdgcn_wmma_f32_16x16x4_f32(
            false, a, false, b1, (short)0, acc1, false, false);
    }

    // C/D layout: N = lane16; VGPR r -> M = r (lanes 0-15), M = r+8 (lanes 16-31)
    const int oCol0 = oBase + lane16;
    const int oCol1 = oCol0 + 16;
    const float bo0 = bias[oCol0];
    const float bo1 = bias[oCol1];
    const int mRow0 = mBase + ((lane >> 4) << 3);
    #pragma unroll
    for (int r = 0; r < 8; ++r) {
        size_t idx0 = (size_t)(mRow0 + r) * DM + oCol0;
        size_t idx1 = (size_t)(mRow0 + r) * DM + oCol1;
        out[idx0] = acc0[r] + bo0 + u[idx0];
        out[idx1] = acc1[r] + bo1 + u[idx1];
    }
}

// ---------- kernel 4: in-place LayerNorm over rows of 512 ----------
__global__ __launch_bounds__(256) void ln_kernel(
    float* __restrict__ r, const float* __restrict__ lnw,
    const float* __restrict__ lnb) {
    __shared__ float ssum[256], ssq[256];
    const int row = blockIdx.x;
    const int tid = threadIdx.x;
    float* rp = r + (size_t)row * DM;
    float x0 = rp[tid], x1 = rp[tid + 256];
    ssum[tid] = x0 + x1;
    ssq[tid]  = x0 * x0 + x1 * x1;
    __syncthreads();
    for (int off = 128; off > 0; off >>= 1) {
        if (tid < off) { ssum[tid] += ssum[tid + off]; ssq[tid] += ssq[tid + off]; }
        __syncthreads();
    }
    const float mean = ssum[0] * (1.0f / (float)DM);
    const float var  = ssq[0] * (1.0f / (float)DM) - mean * mean;
    const float inv  = rsqrtf(var + 1e-5f);
    rp[tid]       = (x0 - mean) * inv * lnw[tid]       + lnb[tid];
    rp[tid + 256] = (x1 - mean) * inv * lnw[tid + 256] + lnb[tid + 256];
}

extern "C" void kernel_launch(void* const* d_in, const int* in_sizes, int n_in,
                              void* d_out, int out_size, void* d_ws, size_t ws_size,
                              hipStream_t stream) {
    const float* u        = (const float*)d_in[0];
    const float* lam_re   = (const float*)d_in[1];
    const float* lam_im   = (const float*)d_in[2];
    const float* p_re     = (const float*)d_in[3];
    const float* p_im     = (const float*)d_in[4];
    const float* B_re     = (const float*)d_in[5];
    const float* B_im     = (const float*)d_in[6];
    const float* Ct_re    = (const float*)d_in[7];
    const float* Ct_im    = (const float*)d_in[8];
    const float* log_step = (const float*)d_in[9];
    const float* Dvec     = (const float*)d_in[10];
    const float* W        = (const float*)d_in[11];
    const float* bias     = (const float*)d_in[12];
    const float* ln_w     = (const float*)d_in[13];
    const float* ln_b     = (const float*)d_in[14];
    float* out = (float*)d_out;

    char* ws = (char*)d_ws;
    float2* Kf  = (float2*)ws;                                        // 16 MB
    float*  zbf = (float*)(ws + (size_t)DM * NFFT * sizeof(float2));  // 32 MB

    kgen_kernel<<<DM, 256, 0, stream>>>(lam_re, lam_im, p_re, p_im,
                                        B_re, B_im, Ct_re, Ct_im, log_step, Kf);
    conv_kernel<<<BATCH * DM, 256, 0, stream>>>(u, Dvec, Kf, zbf);
    const int mTiles = (BATCH * L_SEQ) / 16;   // 1024
    const int nTiles = DM / 32;                // 16
    gemm_kernel<<<(mTiles * nTiles) / 8, 256, 0, stream>>>(zbf, W, bias, u, out);
    ln_kernel<<<BATCH * L_SEQ, 256, 0, stream>>>(out, ln_w, ln_b);
}